// PortfolioGenerator_3152505995652
// MI455X (gfx1250) — compile-verified
//
#include <hip/hip_runtime.h>
#include <stdint.h>

#define NUM_STOCKS 2048
#define WINNER     256
#define THREADS    256
#define PER_T      (NUM_STOCKS / THREADS)   // 8 elements per thread

typedef unsigned int __attribute__((ext_vector_type(4))) v4u;
typedef int          __attribute__((ext_vector_type(4))) v4i;
typedef int          __attribute__((ext_vector_type(8))) v8i;

// Monotone map: float -> uint such that key order == float order.
__device__ __forceinline__ unsigned keyOf(float f) {
    unsigned u = __float_as_uint(f);
    return u ^ (((unsigned)((int)u >> 31)) | 0x80000000u);
}

// Radix-select the G-th largest key (in xorall-transformed key space) over the
// 2048 LDS-resident values. Returns the threshold key; *needed_out = number of
// elements EQUAL to the threshold that belong to the top-G set.
__device__ unsigned radix_select_top(const float* sv, unsigned xorall,
                                     unsigned* hist, unsigned* scan,
                                     unsigned* bshare, unsigned* needed_out) {
    const int tid = threadIdx.x;
    unsigned remaining = WINNER;
    unsigned pref = 0;
    for (int shift = 24; shift >= 0; shift -= 8) {
        hist[tid] = 0;
        __syncthreads();
        unsigned hmask = (shift == 24) ? 0u : (0xFFFFFFFFu << (shift + 8));
#pragma unroll
        for (int i = 0; i < PER_T; ++i) {
            unsigned k = keyOf(sv[tid * PER_T + i]) ^ xorall;
            if ((k & hmask) == (pref & hmask))
                atomicAdd(&hist[(k >> shift) & 0xFFu], 1u);
        }
        __syncthreads();
        // suffix inclusive sum: scan[i] = sum_{j>=i} hist[j]
        scan[tid] = hist[tid];
        __syncthreads();
        for (int off = 1; off < THREADS; off <<= 1) {
            unsigned t = (tid + off < THREADS) ? scan[tid + off] : 0u;
            __syncthreads();
            scan[tid] += t;
            __syncthreads();
        }
        unsigned above = (tid == THREADS - 1) ? 0u : scan[tid + 1];
        if (scan[tid] >= remaining && above < remaining) {
            bshare[0] = (unsigned)tid;   // chosen bucket
            bshare[1] = above;           // count strictly greater
        }
        __syncthreads();
        pref |= bshare[0] << shift;
        remaining -= bshare[1];
        __syncthreads();
    }
    *needed_out = remaining;
    return pref;
}

// Build unnormalized softmax weights in LDS (0 for unselected), return sum.
// negate==0: exp(v - base);  negate==1: exp(base - v).
__device__ float build_weights(const float* sv, unsigned xorall, unsigned thr,
                               unsigned needed, float base, int negate,
                               float* w, unsigned* cnt, float* red) {
    const int tid = threadIdx.x;
    unsigned myeq = 0;
#pragma unroll
    for (int i = 0; i < PER_T; ++i)
        myeq += (keyOf(sv[tid * PER_T + i]) ^ xorall) == thr;
    cnt[tid] = myeq;
    __syncthreads();
    // forward inclusive scan over per-thread equal-counts
    for (int off = 1; off < THREADS; off <<= 1) {
        unsigned t = (tid >= off) ? cnt[tid - off] : 0u;
        __syncthreads();
        cnt[tid] += t;
        __syncthreads();
    }
    unsigned rank = cnt[tid] - myeq;   // exclusive prefix: index-ordered tie rank
    float tsum = 0.f;
#pragma unroll
    for (int i = 0; i < PER_T; ++i) {
        int j = tid * PER_T + i;
        float v = sv[j];
        unsigned k = keyOf(v) ^ xorall;
        bool sel = (k > thr) || (k == thr && rank < needed);
        if (k == thr) rank++;
        float e = sel ? __expf(negate ? (base - v) : (v - base)) : 0.f;
        w[j] = e;
        tsum += e;
    }
    red[tid] = tsum;
    __syncthreads();
    for (int s = THREADS / 2; s > 0; s >>= 1) {
        if (tid < s) red[tid] += red[tid + s];
        __syncthreads();
    }
    float Z = red[0];
    __syncthreads();
    return Z;
}

__global__ __launch_bounds__(THREADS)
void portfolio_kernel(const float* __restrict__ scores,
                      const float* __restrict__ sratio,
                      float* __restrict__ out, int B) {
    __shared__ float    s_vals[NUM_STOCKS];
    __shared__ float    s_wl[NUM_STOCKS];
    __shared__ float    s_ws[NUM_STOCKS];
    __shared__ unsigned s_hist[THREADS];
    __shared__ unsigned s_scan[THREADS];
    __shared__ float    s_red[THREADS];
    __shared__ unsigned s_b[4];

    const int tid = threadIdx.x;
    const int row = blockIdx.x;
    const float* rowp = scores + (size_t)row * NUM_STOCKS;

    // ---- TDM: DMA the whole row (8 KB) global -> LDS; wave 0 only ----------
    if (__builtin_amdgcn_readfirstlane((int)threadIdx.x) < 32) {
        unsigned lds_off = (unsigned)(uintptr_t)(&s_vals[0]);
        unsigned long long ga = (unsigned long long)(uintptr_t)rowp;
        v4u g0;
        g0.x = 1u;                                    // count=1, user descriptor
        g0.y = lds_off;                               // lds_addr
        g0.z = (unsigned)(ga & 0xFFFFFFFFu);          // global_addr[31:0]
        g0.w = ((unsigned)(ga >> 32) & 0x01FFFFFFu)   // global_addr[56:32]
               | 0x80000000u;                         // type=2 ("image")
        v8i g1;
        g1[0] = 0x00020000;                 // data_size=2 -> 4 bytes/elem
        g1[1] = (int)(2048u << 16);         // tensor_dim0[15:0] = 2048
        g1[2] = (int)(1u << 16);            // tensor_dim0 hi=0; tensor_dim1=1
        g1[3] = (int)(2048u << 16);         // tile_dim0 = 2048
        g1[4] = 1;                          // tile_dim1 = 1, tile_dim2 = 0
        g1[5] = 2048;                       // tensor_dim0_stride = 2048
        g1[6] = 0;
        g1[7] = 0;
        v4i g2 = {0, 0, 0, 0};
        v4i g3 = {0, 0, 0, 0};
#if __clang_major__ >= 23
        v8i g4 = {0, 0, 0, 0, 0, 0, 0, 0};
        __builtin_amdgcn_tensor_load_to_lds(g0, g1, g2, g3, g4, 0);
#else
        __builtin_amdgcn_tensor_load_to_lds(g0, g1, g2, g3, 0);
#endif
        __builtin_amdgcn_s_wait_tensorcnt(0);
    }
    __syncthreads();

    // ---- row max / min (softmax stabilizers) -------------------------------
    float mx = s_vals[tid * PER_T], mn = mx;
#pragma unroll
    for (int i = 1; i < PER_T; ++i) {
        float v = s_vals[tid * PER_T + i];
        mx = fmaxf(mx, v);
        mn = fminf(mn, v);
    }
    s_red[tid] = mx;
    __syncthreads();
    for (int s = THREADS / 2; s > 0; s >>= 1) {
        if (tid < s) s_red[tid] = fmaxf(s_red[tid], s_red[tid + s]);
        __syncthreads();
    }
    mx = s_red[0];
    __syncthreads();
    s_red[tid] = mn;
    __syncthreads();
    for (int s = THREADS / 2; s > 0; s >>= 1) {
        if (tid < s) s_red[tid] = fminf(s_red[tid], s_red[tid + s]);
        __syncthreads();
    }
    mn = s_red[0];
    __syncthreads();

    // ---- exact thresholds via radix select ---------------------------------
    unsigned neededW, neededL;
    unsigned thrW = radix_select_top(s_vals, 0u,          s_hist, s_scan, s_b, &neededW);
    unsigned thrL = radix_select_top(s_vals, 0xFFFFFFFFu, s_hist, s_scan, s_b, &neededL);

    // ---- unnormalized softmax weights in LDS -------------------------------
    // long:  softmax(winner_scores)         -> exp(v - mx)
    // short: softmax(1 - loser_scores)      -> exp((1-v) - (1-mn)) = exp(mn - v)
    float Zl = build_weights(s_vals, 0u,          thrW, neededW, mx, 0, s_wl, s_hist, s_red);
    float Zs = build_weights(s_vals, 0xFFFFFFFFu, thrL, neededL, mn, 1, s_ws, s_hist, s_red);

    // ---- stream out (coalesced, non-temporal: 256 MB, no reuse) ------------
    float rl = 1.0f / Zl, rs = 1.0f / Zs;
    float* outL = out + (size_t)row * NUM_STOCKS;
    float* outS = out + (size_t)B * NUM_STOCKS + (size_t)row * NUM_STOCKS;
#pragma unroll
    for (int i = 0; i < PER_T; ++i) {
        int j = i * THREADS + tid;
        __builtin_nontemporal_store(s_wl[j] * rl, &outL[j]);
        __builtin_nontemporal_store(s_ws[j] * rs, &outS[j]);
    }
    if (tid == 0) {
        float r = sratio[row];
        out[(size_t)2 * B * NUM_STOCKS + row] = fminf(fmaxf(r, 0.f), 1.f);
    }
}

extern "C" void kernel_launch(void* const* d_in, const int* in_sizes, int n_in,
                              void* d_out, int out_size, void* d_ws, size_t ws_size,
                              hipStream_t stream) {
    const float* scores = (const float*)d_in[0];   // [B, S] f32
    const float* sratio = (const float*)d_in[1];   // [B]    f32
    float* out = (float*)d_out;                    // [B*S] long | [B*S] short | [B] ratio
    int B = in_sizes[1];
    portfolio_kernel<<<B, THREADS, 0, stream>>>(scores, sratio, out, B);
}